// ESNDriver_66640712565001
// MI455X (gfx1250) — compile-verified
//
#include <hip/hip_runtime.h>
#include <hip/hip_bf16.h>

// ESN step on MI455X (gfx1250):
//   y[i,b]   = sum_k vals[k] * state[b, cols[k]]  for rows[k]==i   (COO SpMM, rows sorted)
//   out[b,i] = 0.6*tanh(y[i,b] + proj[b,i] + 1.6) + 0.4*state[b,i]
//
// L2-bandwidth-bound (NNZ * 512 B of state-row gathers ~ 687 MB from L2).
//  1) transpose state -> stateT [N, B] so gathers are contiguous 512 B rows.
//  2) one block per 16 sorted rows -> contiguous COO slice (binary search).
//  3) segment-sum mapped on V_WMMA_F32_16X16X4_F32 with a one-hot-scaled B.
//  4) COO stream staged into LDS by the Tensor Data Mover (tensor_load_to_lds),
//     double-buffered and overlapped with WMMA compute (TENSORcnt + barrier).
//  5) epilogue uses hardware V_TANH_F32 (branch-free) instead of libm tanhf.

#define RES_DIM   8192
#define BATCH     128
#define LEAK      0.6f
#define BIAS      1.6f
#define COO_CHUNK 1024
#define ROWS_PER_BLOCK 16

typedef __attribute__((ext_vector_type(2))) float        v2f;
typedef __attribute__((ext_vector_type(8))) float        v8f;
typedef __attribute__((ext_vector_type(4))) unsigned int u32x4;
typedef __attribute__((ext_vector_type(4))) int          i32x4;
typedef __attribute__((ext_vector_type(8))) int          i32x8;

__device__ __forceinline__ int imin(int a, int b) { return a < b ? a : b; }

// Branch-free tanh: hardware V_TANH_F32 on gfx1250, else exp-based identity.
__device__ __forceinline__ float fast_tanh(float x) {
#if __has_builtin(__builtin_amdgcn_tanhf)
    return __builtin_amdgcn_tanhf(x);
#else
    const float e = __expf(2.0f * x);      // v_exp_f32, no EXEC divergence
    return 1.0f - 2.0f / (e + 1.0f);       // exact identity; saturates at +-1
#endif
}

// ---------------------------------------------------------------------------
// TDM: 1-D DMA of n 4-byte elements from global to LDS (CDNA5 D#, type=2).
// Group1 packing per ISA 8.4: data_size@[17:16], tensor_dim0@[79:48],
// tensor_dim1@[111:80]=1, tile_dim0@[127:112]=n, tile_dim1@[143:128]=1,
// tensor_dim0_stride@[207:160]=n.
// ---------------------------------------------------------------------------
__device__ __forceinline__ void tdm_load_1d_to_lds(unsigned lds_off,
                                                   const void* gsrc, int n) {
    const unsigned long long ga = (unsigned long long)(uintptr_t)gsrc;
    const unsigned un = (unsigned)n;
    u32x4 g0;
    g0[0] = 1u;                                              // count=1, user D#
    g0[1] = lds_off;                                         // lds_addr (bytes)
    g0[2] = (unsigned)ga;                                    // global_addr[31:0]
    g0[3] = (unsigned)((ga >> 32) & 0x01ffffffu) | (2u << 30); // [56:32]|type=2
    i32x8 g1;
    g1[0] = (int)(2u << 16);                                 // data_size = 4 B
    g1[1] = (int)((un & 0xffffu) << 16);                     // tensor_dim0 lo16
    g1[2] = (int)(((un >> 16) & 0xffffu) | (1u << 16));      // dim0 hi | dim1=1
    g1[3] = (int)((un & 0xffffu) << 16);                     // tile_dim0 = n
    g1[4] = 1;                                               // tile_dim1 = 1
    g1[5] = (int)un;                                         // dim0_stride = n
    g1[6] = 0;
    g1[7] = 0;
    i32x4 z4 = {0, 0, 0, 0};
#if defined(__clang_major__) && (__clang_major__ >= 23)
    i32x8 z8 = {0, 0, 0, 0, 0, 0, 0, 0};
    __builtin_amdgcn_tensor_load_to_lds(g0, g1, z4, z4, z8, 0);
#else
    __builtin_amdgcn_tensor_load_to_lds(g0, g1, z4, z4, 0);
#endif
}

__device__ __forceinline__ unsigned lds_off_of(const void* p) {
    return (unsigned)(uintptr_t)p;   // flat LDS addr: offset lives in [31:0]
}

// ---------------------------------------------------------------------------
// Kernel 1: state [B, N] -> stateT [N, B] via LDS tile transpose.
// ---------------------------------------------------------------------------
__global__ __launch_bounds__(256) void esn_transpose_kernel(
    const float* __restrict__ src, float* __restrict__ dst) {
    __shared__ float tile[32][33];
    const int n0 = blockIdx.x * 32;
    const int b0 = blockIdx.y * 32;
    const int tx = threadIdx.x, ty = threadIdx.y;
#pragma unroll
    for (int j = 0; j < 32; j += 8)
        tile[ty + j][tx] = src[(b0 + ty + j) * RES_DIM + (n0 + tx)];
    __syncthreads();
#pragma unroll
    for (int j = 0; j < 32; j += 8)
        dst[(n0 + ty + j) * BATCH + (b0 + tx)] = tile[tx][ty + j];
}

// ---------------------------------------------------------------------------
// Kernel 2: TDM-staged, double-buffered WMMA SpMM + fused tanh/leak epilogue.
// ---------------------------------------------------------------------------
__global__ __launch_bounds__(256) void esn_spmm_wmma_kernel(
    const float* __restrict__ proj,    // [BATCH, RES_DIM]
    const float* __restrict__ state,   // [BATCH, RES_DIM]
    const float* __restrict__ vals,    // [NNZ]
    const int*   __restrict__ rows,    // [NNZ] sorted
    const int*   __restrict__ cols,    // [NNZ]
    const float* __restrict__ stateT,  // [RES_DIM, BATCH]
    float* __restrict__ out,           // [BATCH, RES_DIM]
    int nnz)
{
    __shared__ __align__(16) float s_vals[2][COO_CHUNK + 8];
    __shared__ __align__(16) int   s_rows[2][COO_CHUNK + 8];
    __shared__ __align__(16) int   s_cols[2][COO_CHUNK + 8];
    __shared__ int s_bounds[2];

    const int rowbase = blockIdx.x * ROWS_PER_BLOCK;
    const int tid     = threadIdx.x;

    if (tid < 2) {
        const int target = rowbase + tid * ROWS_PER_BLOCK;
        int lo = 0, hi = nnz;
        while (lo < hi) {
            int mid = (lo + hi) >> 1;
            if (rows[mid] < target) lo = mid + 1; else hi = mid;
        }
        s_bounds[tid] = lo;
    }
    __syncthreads();
    const int kStart = s_bounds[0];
    const int kEnd   = s_bounds[1];
    const int nChunks = (kEnd - kStart + COO_CHUNK - 1) / COO_CHUNK;

    const int lane   = tid & 31;
    const int waveId = tid >> 5;
    const int m0     = waveId * 16;       // batch tile base
    const int half   = lane >> 4;
    const int laneM  = lane & 15;
    const int rowTest = rowbase + laneM;  // B one-hot test value
    const float* stPtr = stateT + m0 + laneM;

    v8f acc0 = {}, acc1 = {};

    // Prologue: DMA chunk 0 into buffer 0 (TDM ignores EXEC; tid==0 issues once).
    if (tid == 0 && nChunks > 0) {
        const int c0 = imin(COO_CHUNK, kEnd - kStart);
        tdm_load_1d_to_lds(lds_off_of(&s_vals[0][0]), vals + kStart, c0);
        tdm_load_1d_to_lds(lds_off_of(&s_rows[0][0]), rows + kStart, c0);
        tdm_load_1d_to_lds(lds_off_of(&s_cols[0][0]), cols + kStart, c0);
    }

    for (int ci = 0; ci < nChunks; ++ci) {
        const int buf = ci & 1;
        const int k0  = kStart + ci * COO_CHUNK;
        const int cnt = imin(COO_CHUNK, kEnd - k0);

        if (tid == 0) __builtin_amdgcn_s_wait_tensorcnt(0);  // chunk ci landed
        __syncthreads();                                     // visible to all

        // Kick off chunk ci+1 into the other buffer while we compute.
        if (tid == 0 && ci + 1 < nChunks) {
            const int k1 = k0 + COO_CHUNK;
            const int c1 = imin(COO_CHUNK, kEnd - k1);
            tdm_load_1d_to_lds(lds_off_of(&s_vals[buf ^ 1][0]), vals + k1, c1);
            tdm_load_1d_to_lds(lds_off_of(&s_rows[buf ^ 1][0]), rows + k1, c1);
            tdm_load_1d_to_lds(lds_off_of(&s_cols[buf ^ 1][0]), cols + k1, c1);
        }

        // Sentinel-pad the tail once (uniform branch) so the hot loop is
        // branch-free: row=-1 never matches rowTest, col=0 is a safe gather.
        const bool lastChunk = (ci == nChunks - 1);
        if (lastChunk) {
            if (tid < 8) { s_rows[buf][cnt + tid] = -1; s_cols[buf][cnt + tid] = 0; }
            __syncthreads();
        }

        const int cntPad = (cnt + 7) & ~7;
        for (int c = 0; c < cntPad; c += 8) {
            const int kA = c + (half << 1);   // even -> aligned b64 pair loads

            const int2 cp0 = *(const int2*)&s_cols[buf][kA];
            const int2 rp0 = *(const int2*)&s_rows[buf][kA];
            const v2f  vp0 = *(const v2f*) &s_vals[buf][kA];
            v2f a0, b0;
            a0.x = stPtr[cp0.x << 7];
            a0.y = stPtr[cp0.y << 7];
            b0.x = (rp0.x == rowTest) ? vp0.x : 0.0f;
            b0.y = (rp0.y == rowTest) ? vp0.y : 0.0f;
            acc0 = __builtin_amdgcn_wmma_f32_16x16x4_f32(
                false, a0, false, b0, (short)0, acc0, false, false);

            const int2 cp1 = *(const int2*)&s_cols[buf][kA + 4];
            const int2 rp1 = *(const int2*)&s_rows[buf][kA + 4];
            const v2f  vp1 = *(const v2f*) &s_vals[buf][kA + 4];
            v2f a1, b1;
            a1.x = stPtr[cp1.x << 7];
            a1.y = stPtr[cp1.y << 7];
            b1.x = (rp1.x == rowTest) ? vp1.x : 0.0f;
            b1.y = (rp1.y == rowTest) ? vp1.y : 0.0f;
            acc1 = __builtin_amdgcn_wmma_f32_16x16x4_f32(
                false, a1, false, b1, (short)0, acc1, false, false);
        }
        // Top barrier of iteration ci+1 fences these reads against the DMA
        // refill of this buffer (issued at iteration ci+1 for chunk ci+2).
    }

    // Epilogue: C VGPR v: half0 -> M=v, half1 -> M=v+8; N = laneM.
    const v8f acc = acc0 + acc1;
    const int i = rowbase + laneM;
#pragma unroll
    for (int v = 0; v < 8; ++v) {
        const int b   = m0 + v + half * 8;
        const int idx = b * RES_DIM + i;
        const float t = fast_tanh(acc[v] + proj[idx] + BIAS);
        out[idx] = LEAK * t + (1.0f - LEAK) * state[idx];
    }
}

// ---------------------------------------------------------------------------
extern "C" void kernel_launch(void* const* d_in, const int* in_sizes, int n_in,
                              void* d_out, int out_size, void* d_ws, size_t ws_size,
                              hipStream_t stream) {
    const float* proj  = (const float*)d_in[0];
    const float* state = (const float*)d_in[1];
    const float* vals  = (const float*)d_in[2];
    const int*   rows  = (const int*)d_in[3];
    const int*   cols  = (const int*)d_in[4];
    float*       out   = (float*)d_out;
    float*       stateT = (float*)d_ws;          // RES_DIM*BATCH floats = 4 MB
    const int    nnz   = in_sizes[2];

    dim3 tb(32, 8);
    dim3 tg(RES_DIM / 32, BATCH / 32);
    esn_transpose_kernel<<<tg, tb, 0, stream>>>(state, stateT);

    esn_spmm_wmma_kernel<<<RES_DIM / ROWS_PER_BLOCK, 256, 0, stream>>>(
        proj, state, vals, rows, cols, stateT, out, nnz);
}